// SoftTargetLoss_62646392979666
// MI455X (gfx1250) — compile-verified
//
#include <hip/hip_runtime.h>

typedef __attribute__((ext_vector_type(4))) float f32x4;
typedef __attribute__((ext_vector_type(2))) float v2f;
typedef __attribute__((ext_vector_type(8))) float v8f;

#define NUM_CLASSES 512
#define BATCH 65536
#define WAVES_PER_BLOCK 8
#define THREADS (WAVES_PER_BLOCK * 32)

// Full-wave (32-lane) sum using the matrix pipe:
// A (16x4 f32): A[m][0] = v_m (lanes 0-15), A[m][2] = v_{m+16} (lanes 16-31), K=1,3 zero.
// B (4x16)     = all ones  =>  D[m][n] = v_m + v_{m+16}, broadcast across N.
// Each lane then holds D[m][n] for 8 values of m; summing those 8 VGPRs gives the
// half-sum (m=0..7 on lanes 0-15, m=8..15 on lanes 16-31); one shfl_xor(16) completes it.
// Requires EXEC == all ones (call sites are uniform, full blocks only).
__device__ __forceinline__ float wave_sum_wmma(float v) {
    v2f a; a[0] = v;    a[1] = 0.0f;
    v2f b; b[0] = 1.0f; b[1] = 1.0f;
    v8f c = {};
    c = __builtin_amdgcn_wmma_f32_16x16x4_f32(false, a, false, b, (short)0, c, false, false);
    float s = ((c[0] + c[1]) + (c[2] + c[3])) + ((c[4] + c[5]) + (c[6] + c[7]));
    s += __shfl_xor(s, 16, 32);
    return s;
}

__device__ __forceinline__ float wave_max(float v) {
#pragma unroll
    for (int off = 16; off > 0; off >>= 1)
        v = fmaxf(v, __shfl_xor(v, off, 32));
    return v;
}

__global__ void __launch_bounds__(THREADS)
soft_target_loss_main(const float* __restrict__ logits,
                      const int* __restrict__ targets,
                      float* __restrict__ partial) {
    const int lane = threadIdx.x & 31;
    const int wave = threadIdx.x >> 5;
    const int row  = blockIdx.x * WAVES_PER_BLOCK + wave;

    const float* rowp = logits + (size_t)row * NUM_CLASSES;

    // Lane l owns classes {4l..4l+3} + 128k : each b128 load is a contiguous 512B span.
    f32x4 x[4];
#pragma unroll
    for (int k = 0; k < 4; ++k)
        x[k] = __builtin_nontemporal_load((const f32x4*)(rowp + 128 * k + 4 * lane));

    // Row max (streaming data is N(0,1); still do it properly).
    float m = x[0][0];
#pragma unroll
    for (int k = 0; k < 4; ++k)
#pragma unroll
        for (int i = 0; i < 4; ++i)
            m = fmaxf(m, x[k][i]);
    m = wave_max(m);

    // Sum of exp(x - m), lane-local then WMMA wave sum.
    float s = 0.0f;
#pragma unroll
    for (int k = 0; k < 4; ++k)
#pragma unroll
        for (int i = 0; i < 4; ++i)
            s += __expf(x[k][i] - m);
    const float S    = wave_sum_wmma(s);
    const float logZ = m + __logf(S);

    // Triangular soft-target window around c, clipped at [0, C).
    const int c = targets[row];
    float W = 0.0f;
#pragma unroll
    for (int dj = -2; dj <= 2; ++dj) {
        const int j = c + dj;
        if (j >= 0 && j < NUM_CLASSES)
            W += (float)(3 - (dj < 0 ? -dj : dj));
    }
    const float invW = 1.0f / W;

    // KL contribution: only the (<=5) classes inside the window are non-zero.
    float acc = 0.0f;
#pragma unroll
    for (int k = 0; k < 4; ++k)
#pragma unroll
        for (int i = 0; i < 4; ++i) {
            const int j = 128 * k + 4 * lane + i;
            int d = j - c;
            d = d < 0 ? -d : d;
            if (d <= 2) {
                const float st = (float)(3 - d) * invW;
                acc += st * (__logf(st) - (x[k][i] - logZ));
            }
        }
    const float kl = wave_sum_wmma(acc);

    __shared__ float smem[WAVES_PER_BLOCK];
    if (lane == 0) smem[wave] = kl;
    __syncthreads();
    if (threadIdx.x == 0) {
        float t = 0.0f;
#pragma unroll
        for (int w = 0; w < WAVES_PER_BLOCK; ++w) t += smem[w];
        partial[blockIdx.x] = t;
    }
}

__global__ void __launch_bounds__(256)
reduce_partials(const float* __restrict__ partial, int n,
                float* __restrict__ out, float scale) {
    float s = 0.0f;
    // n is a multiple of 256 -> loop trip count uniform, EXEC stays all-ones.
    for (int i = threadIdx.x; i < n; i += 256)
        s += partial[i];
    s = wave_sum_wmma(s);

    __shared__ float smem[8];
    if ((threadIdx.x & 31) == 0) smem[threadIdx.x >> 5] = s;
    __syncthreads();
    if (threadIdx.x == 0) {
        float t = 0.0f;
#pragma unroll
        for (int w = 0; w < 8; ++w) t += smem[w];
        out[0] = t * scale;
    }
}

extern "C" void kernel_launch(void* const* d_in, const int* in_sizes, int n_in,
                              void* d_out, int out_size, void* d_ws, size_t ws_size,
                              hipStream_t stream) {
    const float* logits  = (const float*)d_in[0];
    const int*   targets = (const int*)d_in[1];
    float*       out     = (float*)d_out;
    float*       partial = (float*)d_ws;   // 8192 floats = 32 KB of scratch

    const int nblocks = BATCH / WAVES_PER_BLOCK;  // 8192 blocks, 1 wave per row
    soft_target_loss_main<<<nblocks, THREADS, 0, stream>>>(logits, targets, partial);
    reduce_partials<<<1, 256, 0, stream>>>(partial, nblocks, out, 1.0f / (float)BATCH);
}